// SmallerPredAlexNet_39135742001967
// MI455X (gfx1250) — compile-verified
//
#include <hip/hip_runtime.h>
#include <hip/hip_bf16.h>

typedef float v2f __attribute__((ext_vector_type(2)));
typedef float v8f __attribute__((ext_vector_type(8)));

// ---------------------------------------------------------------------------
// Generic M=32 GEMM on the fp32 WMMA path:  C[m, j] = sum_k A[m,k] * W[wRow0+j, k]
// A: [32, K] row-major (K multiple of 4).  W: rows of length K (row-major).
// C: [32, Ncols] row-major (Ncols multiple of 16). Optional bias added per column.
// One wave handles one 16-column tile and both 16-row halves (M=32).
// fp32 WMMA 16x16x4 layouts (CDNA5 ISA 7.12.2):
//   A (16x4): lane l, m=l&15, g=l>>4 -> vgpr v holds K = 2g+v
//   B (4x16): lane l, n=l&15, g=l>>4 -> vgpr v holds K = 2g+v  (B[k][n] = W[n][k])
//   C (16x16): lane l, n=l&15, g=l>>4 -> vgpr v holds row m = v + 8g
// ---------------------------------------------------------------------------
__global__ void hypergemm_wmma(const float* __restrict__ A, int K,
                               const float* __restrict__ Wt, long long wRow0,
                               const float* __restrict__ bias,
                               float* __restrict__ C, int Ncols)
{
    const int wave   = threadIdx.x >> 5;
    const int ntiles = Ncols >> 4;
    const int tile   = blockIdx.x * (blockDim.x >> 5) + wave;
    if (tile >= ntiles) return;                 // wave-uniform: EXEC stays all-1s

    const int l  = threadIdx.x & 31;
    const int g  = l >> 4;
    const int mr = l & 15;
    const int j0 = tile << 4;

    const float* __restrict__ wp  = Wt + (wRow0 + (long long)(j0 + mr)) * (long long)K;
    const float* __restrict__ a0p = A + (long long)mr * K;
    const float* __restrict__ a1p = A + (long long)(mr + 16) * K;

    v8f c0 = {};
    v8f c1 = {};
    for (int k0 = 0; k0 < K; k0 += 4) {
        const int kA = k0 + 2 * g;
        v2f a0; a0.x = a0p[kA]; a0.y = a0p[kA + 1];
        v2f a1; a1.x = a1p[kA]; a1.y = a1p[kA + 1];
        v2f bm; bm.x = wp[kA];  bm.y = wp[kA + 1];
        c0 = __builtin_amdgcn_wmma_f32_16x16x4_f32(false, a0, false, bm, (short)0, c0, false, false);
        c1 = __builtin_amdgcn_wmma_f32_16x16x4_f32(false, a1, false, bm, (short)0, c1, false, false);
    }

    const float bv = bias ? bias[j0 + mr] : 0.0f;
#pragma unroll
    for (int v = 0; v < 8; ++v) {
        const int row = v + 8 * g;
        C[(long long)row        * Ncols + j0 + mr] = c0[v] + bv;
        C[(long long)(row + 16) * Ncols + j0 + mr] = c1[v] + bv;
    }
}

// ---------------------------------------------------------------------------
// Implicit-GEMM 3x3 conv (pad=1, stride=1) on the fp32 WMMA path.
// Requires Cin % 4 == 0 (layers 2..5).  Per block: one sample b, one 16-wide
// output-channel tile.  Input channels processed in chunks of CIC staged into
// LDS together with the 16 generated weight rows for that chunk.
// Each wave owns up to two 16-pixel output tiles:
//   C[co, pix] += A[co, k] * B[k, pix],  A from LDS weights, B = im2col gather
// Fused eval-mode BN2d ((x - m)/sqrt(1+eps)) and ReLU on store.
// ---------------------------------------------------------------------------
__device__ __forceinline__ float im2col_val(const float* __restrict__ sIn,
                                            int HW, int H, int W,
                                            int oh, int ow, int ci, int kh, int kw,
                                            bool valid)
{
    const int ih = oh + kh - 1;
    const int iw = ow + kw - 1;
    const bool ok = valid & (ih >= 0) & (ih < H) & (iw >= 0) & (iw < W);
    const float v = sIn[ci * HW + ih * W + iw];
    return ok ? v : 0.0f;
}

__global__ void conv3x3_wmma(const float* __restrict__ in, int Cin, int H, int W,
                             const float* __restrict__ cw, int chunkN, int CIC,
                             int coStart, int CoutTotal,
                             float* __restrict__ out,
                             const float* __restrict__ bn_m, int doRelu)
{
    __shared__ float sIn[5400];       // max CIC*HW  (24*225)
    __shared__ float sW[16 * 577];    // 16 rows, stride Kc+1 (bank-conflict pad)

    const int b   = blockIdx.y;
    const int coT = blockIdx.x;
    const int HW  = H * W;
    const int NT  = (HW + 15) >> 4;   // 16-pixel output tiles
    const int nw  = blockDim.x >> 5;
    const int wv  = threadIdx.x >> 5;
    const int l   = threadIdx.x & 31;
    const int g   = l >> 4;
    const int mr  = l & 15;
    const int K   = Cin * 9;
    const int Kc  = CIC * 9;
    const int KcP = Kc + 1;

    const int  t0 = wv, t1 = wv + nw;
    const bool w0 = (t0 < NT), w1 = (t1 < NT);   // wave-uniform

    const int  pix0 = t0 * 16 + mr;  const bool vp0 = pix0 < HW;
    const int  oh0  = pix0 / W;      const int  ow0 = pix0 - oh0 * W;
    const int  pix1 = t1 * 16 + mr;  const bool vp1 = pix1 < HW;
    const int  oh1  = pix1 / W;      const int  ow1 = pix1 - oh1 * W;

    v8f acc0 = {};
    v8f acc1 = {};
    const float* __restrict__ cwb = cw + (long long)b * chunkN + (long long)(coT * 16) * K;

    for (int cic = 0; cic < Cin; cic += CIC) {
        __syncthreads();   // protect LDS reuse (WAR with previous chunk's reads)
        for (int i = threadIdx.x; i < CIC * HW; i += blockDim.x)
            sIn[i] = in[((long long)b * Cin + cic) * HW + i];
        for (int i = threadIdx.x; i < 16 * Kc; i += blockDim.x) {
            const int coL = i / Kc;
            const int kk  = i - coL * Kc;
            sW[coL * KcP + kk] = cwb[(long long)coL * K + cic * 9 + kk];
        }
        __syncthreads();

        if (w0) {                                  // wave-uniform guard
            for (int k0 = 0; k0 < Kc; k0 += 4) {
                const int kA = k0 + 2 * g;
                v2f a; a.x = sW[mr * KcP + kA]; a.y = sW[mr * KcP + kA + 1];

                const int ciA = kA / 9, rA = kA - 9 * ciA;
                const int khA = rA / 3, kwA = rA - 3 * khA;
                const int kB  = kA + 1;
                const int ciB = kB / 9, rB = kB - 9 * ciB;
                const int khB = rB / 3, kwB = rB - 3 * khB;

                v2f bb0;
                bb0.x = im2col_val(sIn, HW, H, W, oh0, ow0, ciA, khA, kwA, vp0);
                bb0.y = im2col_val(sIn, HW, H, W, oh0, ow0, ciB, khB, kwB, vp0);
                acc0 = __builtin_amdgcn_wmma_f32_16x16x4_f32(false, a, false, bb0, (short)0, acc0, false, false);

                if (w1) {                          // wave-uniform guard
                    v2f bb1;
                    bb1.x = im2col_val(sIn, HW, H, W, oh1, ow1, ciA, khA, kwA, vp1);
                    bb1.y = im2col_val(sIn, HW, H, W, oh1, ow1, ciB, khB, kwB, vp1);
                    acc1 = __builtin_amdgcn_wmma_f32_16x16x4_f32(false, a, false, bb1, (short)0, acc1, false, false);
                }
            }
        }
    }

    const float inv = 0.9999950000374997f;        // 1/sqrt(1 + 1e-5)
    auto store_tile = [&](const v8f& acc, int oh, int ow, bool vp) {
        if (!vp) return;
#pragma unroll
        for (int v = 0; v < 8; ++v) {
            const int co = coStart + coT * 16 + v + 8 * g;
            float val = acc[v];
            if (bn_m)   val = (val - bn_m[co]) * inv;
            if (doRelu) val = fmaxf(val, 0.0f);
            out[(((long long)b * CoutTotal + co) * H + oh) * W + ow] = val;
        }
    };
    if (w0) store_tile(acc0, oh0, ow0, vp0);
    if (w1) store_tile(acc1, oh1, ow1, vp1);
}

// ---------------------------------------------------------------------------
// Direct VALU conv for layer 1 only (Cin=3, K=27 not a multiple of 4).
// ---------------------------------------------------------------------------
__global__ void conv3x3_fused(const float* __restrict__ in, int Cin, int H, int W,
                              const float* __restrict__ cw, int chunkN,
                              int coStart, int CoutTotal,
                              float* __restrict__ out,
                              const float* __restrict__ bn_m, int doRelu)
{
    __shared__ float wl[64];
    const int coL = blockIdx.x;
    const int b   = blockIdx.y;
    const int K   = Cin * 9;

    const float* __restrict__ wsrc = cw + (long long)b * chunkN + (long long)coL * K;
    for (int k = threadIdx.x; k < K; k += blockDim.x) wl[k] = wsrc[k];
    __syncthreads();

    const int   HW  = H * W;
    const int   co  = coStart + coL;
    const float inv = 0.9999950000374997f;
    const float m   = bn_m ? bn_m[co] : 0.0f;
    const float* __restrict__ ib = in + (long long)b * Cin * HW;

    for (int pix = threadIdx.x; pix < HW; pix += blockDim.x) {
        const int oh = pix / W;
        const int ow = pix - oh * W;
        float acc = 0.0f;
        for (int ci = 0; ci < Cin; ++ci) {
            const float* __restrict__ ic = ib + (long long)ci * HW;
#pragma unroll
            for (int kh = 0; kh < 3; ++kh) {
                const int ih = oh + kh - 1;
                if (ih < 0 || ih >= H) continue;
#pragma unroll
                for (int kw = 0; kw < 3; ++kw) {
                    const int iw = ow + kw - 1;
                    if (iw < 0 || iw >= W) continue;
                    acc = fmaf(wl[ci * 9 + kh * 3 + kw], ic[ih * W + iw], acc);
                }
            }
        }
        if (bn_m)   acc = (acc - m) * inv;
        if (doRelu) acc = fmaxf(acc, 0.0f);
        out[(((long long)b * CoutTotal + co) * H + oh) * W + ow] = acc;
    }
}

// MaxPool 3x3, stride 2, VALID.
__global__ void maxpool3s2(const float* __restrict__ in, int C, int Hi, int Wi,
                           float* __restrict__ out, int Ho, int Wo)
{
    const int total = 32 * C * Ho * Wo;
    const int idx = blockIdx.x * blockDim.x + threadIdx.x;
    if (idx >= total) return;
    int t  = idx;
    const int wo = t % Wo; t /= Wo;
    const int ho = t % Ho; t /= Ho;
    const int c  = t % C;
    const int b  = t / C;
    const float* __restrict__ ip = in + ((long long)(b * C + c) * Hi) * Wi;
    float mx = -3.402823466e+38f;
#pragma unroll
    for (int kh = 0; kh < 3; ++kh)
#pragma unroll
        for (int kw = 0; kw < 3; ++kw)
            mx = fmaxf(mx, ip[(ho * 2 + kh) * Wi + (wo * 2 + kw)]);
    out[idx] = mx;
}

// Concatenate flattened h5 [32,1728] with y [32,32] -> hcat [32,1760].
__global__ void build_hcat(const float* __restrict__ h5, const float* __restrict__ y,
                           float* __restrict__ hcat)
{
    const int idx = blockIdx.x * blockDim.x + threadIdx.x;
    if (idx >= 32 * 1760) return;
    const int b = idx / 1760;
    const int j = idx - b * 1760;
    hcat[idx] = (j < 1728) ? h5[b * 1728 + j] : y[b * 32 + (j - 1728)];
}

// Training-mode BatchNorm1d (batch stats over B=32, biased var) + affine + ReLU.
__global__ void bn1d_relu(const float* __restrict__ z, const float* __restrict__ g,
                          const float* __restrict__ b, float* __restrict__ out, int N)
{
    const int j = blockIdx.x * blockDim.x + threadIdx.x;
    if (j >= N) return;
    float mu = 0.0f;
    for (int i = 0; i < 32; ++i) mu += z[i * N + j];
    mu *= (1.0f / 32.0f);
    float var = 0.0f;
    for (int i = 0; i < 32; ++i) { const float d = z[i * N + j] - mu; var = fmaf(d, d, var); }
    var *= (1.0f / 32.0f);
    const float s = g[j] * rsqrtf(var + 1e-5f);
    const float o = b[j] - mu * s;
    for (int i = 0; i < 32; ++i) out[i * N + j] = fmaxf(fmaf(z[i * N + j], s, o), 0.0f);
}

// Row-wise L2 normalize of z [32,128] -> out.
__global__ void l2norm_rows(const float* __restrict__ z, float* __restrict__ out)
{
    __shared__ float sq[128];
    const int b = blockIdx.x;
    const int t = threadIdx.x;
    const float v = z[b * 128 + t];
    sq[t] = v * v;
    __syncthreads();
    for (int s = 64; s > 0; s >>= 1) {
        if (t < s) sq[t] += sq[t + s];
        __syncthreads();
    }
    const float n = fmaxf(sqrtf(sq[0]), 1e-12f);
    out[b * 128 + t] = v / n;
}

// ---------------------------------------------------------------------------
extern "C" void kernel_launch(void* const* d_in, const int* in_sizes, int n_in,
                              void* d_out, int out_size, void* d_ws, size_t ws_size,
                              hipStream_t stream)
{
    const float* x    = (const float*)d_in[0];
    const float* y    = (const float*)d_in[1];
    const float* W1   = (const float*)d_in[2];
    const float* W2   = (const float*)d_in[3];
    const float* W3   = (const float*)d_in[4];
    const float* W4   = (const float*)d_in[5];
    const float* W5   = (const float*)d_in[6];
    const float* bnm1 = (const float*)d_in[7];
    const float* bnm2 = (const float*)d_in[8];
    const float* bnm3 = (const float*)d_in[9];
    const float* bnm4 = (const float*)d_in[10];
    const float* Wb0  = (const float*)d_in[11];
    const float* g0   = (const float*)d_in[12];
    const float* b0   = (const float*)d_in[13];
    const float* Wb1  = (const float*)d_in[14];
    const float* g1   = (const float*)d_in[15];
    const float* b1   = (const float*)d_in[16];
    const float* Wb2  = (const float*)d_in[17];
    const float* bb2  = (const float*)d_in[18];

    float* ws = (float*)d_ws;
    size_t o = 0;
    float* cwbuf = ws + o; o += 7077888;            // max chunk: 32 * 64 * 3456
    float* h1    = ws + o; o += 32 * 96  * 32 * 32;
    float* p1    = ws + o; o += 32 * 96  * 15 * 15;
    float* h2    = ws + o; o += 32 * 192 * 15 * 15;
    float* p2    = ws + o; o += 32 * 192 * 7  * 7;
    float* h3    = ws + o; o += 32 * 384 * 7  * 7;
    float* h4    = ws + o; o += 32 * 384 * 7  * 7;
    float* p4    = ws + o; o += 32 * 384 * 3  * 3;
    float* h5    = ws + o; o += 32 * 192 * 3  * 3;
    float* hcat  = ws + o; o += 32 * 1760;
    float* z0    = ws + o; o += 32 * 256;
    float* a0    = ws + o; o += 32 * 256;
    float* z1    = ws + o; o += 32 * 256;
    float* a1    = ws + o; o += 32 * 256;
    float* z2    = ws + o; o += 32 * 128;

    auto gen = [&](const float* Wl, long long row0, int chunkN) {
        const int tiles = chunkN / 16;
        hypergemm_wmma<<<dim3((tiles + 7) / 8), 256, 0, stream>>>(
            y, 32, Wl, row0, nullptr, cwbuf, chunkN);
    };
    auto convw = [&](const float* in, int Cin, int H, int W, int chunkN, int CIC,
                     int coChunk, int coStart, int CoutTotal,
                     float* out, const float* bnm, int relu, int threads) {
        conv3x3_wmma<<<dim3(coChunk / 16, 32), threads, 0, stream>>>(
            in, Cin, H, W, cwbuf, chunkN, CIC, coStart, CoutTotal, out, bnm, relu);
    };
    auto pool = [&](const float* in, int C, int Hi, int Wi, float* out, int Ho, int Wo) {
        const int total = 32 * C * Ho * Wo;
        maxpool3s2<<<(total + 255) / 256, 256, 0, stream>>>(in, C, Hi, Wi, out, Ho, Wo);
    };

    // Layer 1 (VALU conv, Cin=3): [32,3,32,32] -> [32,96,32,32]; BN+ReLU; pool -> 15x15
    gen(W1, 0, 96 * 27);
    conv3x3_fused<<<dim3(96, 32), 256, 0, stream>>>(x, 3, 32, 32, cwbuf, 96 * 27,
                                                    0, 96, h1, bnm1, 1);
    pool(h1, 96, 32, 32, p1, 15, 15);

    // Layer 2 (WMMA conv): -> [32,192,15,15]; BN+ReLU; pool -> 7x7
    // NT=15 pixel tiles -> 8 waves x 2 tiles; CIC=24 (4 chunks of 96).
    gen(W2, 0, 192 * 96 * 9);
    convw(p1, 96, 15, 15, 192 * 96 * 9, 24, 192, 0, 192, h2, bnm2, 1, 256);
    pool(h2, 192, 15, 15, p2, 7, 7);

    // Layer 3 (WMMA conv): -> [32,384,7,7]; BN+ReLU. 4 co-chunks of 96; CIC=64.
    for (int c = 0; c < 4; ++c) {
        const int chunkN = 96 * 192 * 9;
        gen(W3, (long long)c * chunkN, chunkN);
        convw(p2, 192, 7, 7, chunkN, 64, 96, c * 96, 384, h3, bnm3, 1, 128);
    }

    // Layer 4 (WMMA conv): -> [32,384,7,7]; BN+ReLU; pool -> 3x3. 6 chunks of 64.
    for (int c = 0; c < 6; ++c) {
        const int chunkN = 64 * 384 * 9;
        gen(W4, (long long)c * chunkN, chunkN);
        convw(h3, 384, 7, 7, chunkN, 64, 64, c * 64, 384, h4, bnm4, 1, 128);
    }
    pool(h4, 384, 7, 7, p4, 3, 3);

    // Layer 5 (WMMA conv): -> [32,192,3,3] raw. 4 chunks of 48; CIC=64.
    for (int c = 0; c < 4; ++c) {
        const int chunkN = 48 * 384 * 9;
        gen(W5, (long long)c * chunkN, chunkN);
        convw(p4, 384, 3, 3, chunkN, 64, 48, c * 48, 192, h5, nullptr, 0, 64);
    }

    // Head
    build_hcat<<<(32 * 1760 + 255) / 256, 256, 0, stream>>>(h5, y, hcat);

    hypergemm_wmma<<<dim3(2), 256, 0, stream>>>(hcat, 1760, Wb0, 0, nullptr, z0, 256);
    bn1d_relu<<<1, 256, 0, stream>>>(z0, g0, b0, a0, 256);

    hypergemm_wmma<<<dim3(2), 256, 0, stream>>>(a0, 256, Wb1, 0, nullptr, z1, 256);
    bn1d_relu<<<1, 256, 0, stream>>>(z1, g1, b1, a1, 256);

    hypergemm_wmma<<<dim3(1), 256, 0, stream>>>(a1, 256, Wb2, 0, bb2, z2, 128);

    l2norm_rows<<<32, 128, 0, stream>>>(z2, (float*)d_out);
}